// ModelNew_85710367359300
// MI455X (gfx1250) — compile-verified
//
#include <hip/hip_runtime.h>

typedef __attribute__((ext_vector_type(2))) float v2f;
typedef __attribute__((ext_vector_type(8))) float v8f;

#define N_BATCH 2
#define N_SEQ   4096
#define N_HEADS 16
#define DH      64     // p
#define DSTATE  128    // n
#define LCHUNK  64     // l (block length) == K of the GEMM

#define XPAD    (DH + 4)      // 68 floats = 272B, 16B-aligned rows
#define BPAD    (DSTATE + 4)  // 132 floats = 528B, 16B-aligned rows

__global__ __launch_bounds__(256) void ssd_last_chunk_wmma(
    const float* __restrict__ X,
    const float* __restrict__ A,
    const float* __restrict__ B,
    float* __restrict__ Out)
{
    __shared__ float sXw[LCHUNK][XPAD];   // weighted X: [l][p]
    __shared__ float sB [LCHUNK][BPAD];   // B:          [l][n]
    __shared__ float sA [LCHUNK];
    __shared__ float sCum[LCHUNK];
    __shared__ float sW [LCHUNK];

    const int bh  = blockIdx.x;
    const int b   = bh / N_HEADS;
    const int h   = bh % N_HEADS;
    const int t0  = N_SEQ - LCHUNK;     // only the last chunk matters (see analysis)
    const int tid = threadIdx.x;

    // ---- A for last chunk ----
    if (tid < LCHUNK) {
        sA[tid] = A[(size_t)(b * N_SEQ + t0 + tid) * N_HEADS + h];
    }
    __syncthreads();

    // serial fp32 inclusive cumsum (matches jnp.cumsum ordering)
    if (tid < LCHUNK) {
        float s = 0.0f;
        for (int j = 0; j <= tid; ++j) s += sA[j];
        sCum[tid] = s;
    }
    __syncthreads();

    // combined weight: exp(A_last) * exp(cum[63] - cum[l]) = exp(2*S - cum[l])
    if (tid < LCHUNK) {
        const float S = sCum[LCHUNK - 1];
        sW[tid] = expf(2.0f * S - sCum[tid]);
    }
    __syncthreads();

    // ---- stage weighted X and B into LDS (float4 global loads, b128 LDS stores) ----
    for (int idx = tid; idx < LCHUNK * (DH / 4); idx += 256) {
        const int l = idx >> 4;          // row (timestep in chunk)
        const int q = idx & 15;          // float4 column
        const float4 v = *(const float4*)(
            X + (((size_t)(b * N_SEQ + t0 + l) * N_HEADS + h) * DH + q * 4));
        const float w = sW[l];
        float4 o; o.x = v.x * w; o.y = v.y * w; o.z = v.z * w; o.w = v.w * w;
        *(float4*)&sXw[l][q * 4] = o;
    }
    for (int idx = tid; idx < LCHUNK * (DSTATE / 4); idx += 256) {
        const int l = idx >> 5;
        const int q = idx & 31;
        const float4 v = *(const float4*)(
            B + (((size_t)(b * N_SEQ + t0 + l) * N_HEADS + h) * DSTATE + q * 4));
        *(float4*)&sB[l][q * 4] = v;
    }
    __syncthreads();

    // ---- fp32 WMMA: out[p,n] = sum_k Xw[k,p] * B[k,n] ----
    const int wave = tid >> 5;           // 0..7 -> n-tile
    const int lane = tid & 31;
    const int half = lane >> 4;          // 0: K pair {0,1} / rows M..; 1: K pair {2,3}
    const int m    = lane & 15;
    const int nb   = wave * 16;

    const v8f vzero = {0.f, 0.f, 0.f, 0.f, 0.f, 0.f, 0.f, 0.f};
    v8f acc[4];
    acc[0] = vzero; acc[1] = vzero; acc[2] = vzero; acc[3] = vzero;

#pragma unroll
    for (int k0 = 0; k0 < LCHUNK; k0 += 4) {
        const int kb = k0 + half * 2;    // lanes 0-15: K=k0,k0+1 ; lanes 16-31: K=k0+2,k0+3
        v2f bf;
        bf.x = sB[kb][nb + m];
        bf.y = sB[kb + 1][nb + m];
#pragma unroll
        for (int pt = 0; pt < 4; ++pt) {
            v2f af;
            af.x = sXw[kb][pt * 16 + m];
            af.y = sXw[kb + 1][pt * 16 + m];
            // 8 args: (neg_a, A, neg_b, B, c_mod, C, reuse_a, reuse_b)
            acc[pt] = __builtin_amdgcn_wmma_f32_16x16x4_f32(
                false, af, false, bf, (short)0, acc[pt], false, false);
        }
    }

    // ---- write D: VGPR r -> M = r + 8*half, N = m (per ISA 16x16 f32 C/D layout) ----
    float* outp = Out + (size_t)(b * N_HEADS + h) * DH * DSTATE;
#pragma unroll
    for (int pt = 0; pt < 4; ++pt) {
#pragma unroll
        for (int r = 0; r < 8; ++r) {
            const int p = pt * 16 + r + half * 8;
            const int n = nb + m;
            outp[(size_t)p * DSTATE + n] = acc[pt][r];
        }
    }
}

extern "C" void kernel_launch(void* const* d_in, const int* in_sizes, int n_in,
                              void* d_out, int out_size, void* d_ws, size_t ws_size,
                              hipStream_t stream) {
    (void)in_sizes; (void)n_in; (void)out_size; (void)d_ws; (void)ws_size;
    const float* X = (const float*)d_in[0];   // (2, 4096, 16, 64)  fp32
    const float* A = (const float*)d_in[1];   // (2, 4096, 16)      fp32
    const float* B = (const float*)d_in[2];   // (2, 4096, 16, 128) fp32
    float* Out = (float*)d_out;               // (2, 16, 64, 128)   fp32

    dim3 grid(N_BATCH * N_HEADS);             // 32 workgroups, one per (b,h)
    dim3 block(256);                          // 8 wave32
    ssd_last_chunk_wmma<<<grid, block, 0, stream>>>(X, A, B, Out);
}